// NearestNeighbor_3779571221027
// MI455X (gfx1250) — compile-verified
//
#include <hip/hip_runtime.h>
#include <hip/hip_bf16.h>
#include <stdint.h>

// NN search: dist_i = sum_k (train[i][k] - q[k])^2 via V_WMMA_F32_16X16X4_F32
// diagonal trick (D = A*A^T, diag = squared distances for 16 candidates/wave).
// Bandwidth-bound: 614 MB single-pass NT stream, ~26 us floor @ 23.3 TB/s.

typedef __attribute__((ext_vector_type(2))) float v2f;
typedef __attribute__((ext_vector_type(8))) float v8f;

#define NCAND 100000
#define OBSF  1536          // 16 obs frames * 96 feat
#define PREDF 2400          // 25 pred frames * 96 feat
#define CAND_PER_WAVE 16
#define WAVES_PER_BLOCK 8
#define CAND_PER_BLOCK (CAND_PER_WAVE * WAVES_PER_BLOCK)   // 128

__global__ void nn_init_kernel(unsigned long long* __restrict__ best) {
    *best = 0xFFFFFFFFFFFFFFFFull;
}

__global__ void __launch_bounds__(256)
nn_dist_kernel(const float* __restrict__ q,          // [OBSF] query pose
               const float* __restrict__ train,      // [NCAND][OBSF]
               unsigned long long* __restrict__ best) // packed (dist_bits<<32)|idx
{
    __shared__ float qs[OBSF];                       // 6 KB of the 320 KB WGP LDS
    const int t = threadIdx.x;
    #pragma unroll
    for (int j = 0; j < OBSF / 256; ++j)
        qs[t + j * 256] = q[t + j * 256];
    __syncthreads();

    const int lane = t & 31;
    const int wave = t >> 5;
    const int base = blockIdx.x * CAND_PER_BLOCK + wave * CAND_PER_WAVE;
    if (base >= NCAND) return;                       // wave-uniform; EXEC stays all-1s

    // A-fragment (16x4 f32): lane m (0-15) holds A[m][k0..k0+1],
    // lane m+16 holds A[m][k0+2..k0+3]  -> lane pair reads a contiguous float2.
    const int m     = lane & 15;
    const int khalf = (lane >> 4) << 1;              // 0 or 2
    const float* rowp = train + (size_t)(base + m) * OBSF + khalf;

    v8f acc = {0.f, 0.f, 0.f, 0.f, 0.f, 0.f, 0.f, 0.f};

    #pragma unroll 8
    for (int k0 = 0; k0 < OBSF; k0 += 4) {
        // single-use 614MB stream: non-temporal so it doesn't churn L2
        v2f tv = __builtin_nontemporal_load((const v2f*)(rowp + k0));
        v2f qv;
        qv.x = qs[k0 + khalf];
        qv.y = qs[k0 + khalf + 1];
        v2f a = tv - qv;                             // diff
        // D = diff_block * diff_block^T + C ; diagonal accumulates sum(diff^2)
        acc = __builtin_amdgcn_wmma_f32_16x16x4_f32(
                  false, a, false, a, (short)0, acc, false, false);
    }

    // Diagonal extraction (C/D layout): lanes 0-7 hold diag[0..7] at acc[lane],
    // lanes 24-31 hold diag[8..15] at acc[lane-24]. Lanes 8-23 hold none.
    const int r = lane & 7;
    float d = acc[0];
    d = (r == 1) ? acc[1] : d;
    d = (r == 2) ? acc[2] : d;
    d = (r == 3) ? acc[3] : d;
    d = (r == 4) ? acc[4] : d;
    d = (r == 5) ? acc[5] : d;
    d = (r == 6) ? acc[6] : d;
    d = (r == 7) ? acc[7] : d;

    const bool valid = (lane < 8) || (lane >= 24);
    const int  cand  = base + ((lane < 8) ? lane : (lane - 16));

    // dist >= 0 -> IEEE bit pattern is monotone as u32; low 32 bits = index
    // so min(key) == (min dist, lowest index) matching jnp.argmin tie-break.
    unsigned long long key =
        ((unsigned long long)__float_as_uint(d) << 32) | (unsigned int)cand;
    if (!valid) key = 0xFFFFFFFFFFFFFFFFull;

    #pragma unroll
    for (int off = 16; off > 0; off >>= 1) {
        unsigned long long o = __shfl_xor(key, off, 32);
        key = (o < key) ? o : key;
    }
    if (lane == 0) atomicMin(best, key);             // one atomic per wave (6250 total)
}

__global__ void nn_gather_kernel(const unsigned long long* __restrict__ best,
                                 const float* __restrict__ target_vels,  // [NCAND][PREDF]
                                 float* __restrict__ out)                // [PREDF]
{
    const unsigned int idx = (unsigned int)(*best & 0xFFFFFFFFull);
    const int i = blockIdx.x * blockDim.x + threadIdx.x;
    if (i < PREDF) out[i] = target_vels[(size_t)idx * PREDF + i];
}

extern "C" void kernel_launch(void* const* d_in, const int* in_sizes, int n_in,
                              void* d_out, int out_size, void* d_ws, size_t ws_size,
                              hipStream_t stream) {
    const float* in_pose     = (const float*)d_in[0];   // [16,96]
    const float* train_poses = (const float*)d_in[1];   // [100000,16,96]
    const float* target_vels = (const float*)d_in[2];   // [100000,25,96]
    float* out = (float*)d_out;                         // [25,96]
    unsigned long long* best = (unsigned long long*)d_ws;

    nn_init_kernel<<<1, 1, 0, stream>>>(best);

    const int blocks = (NCAND + CAND_PER_BLOCK - 1) / CAND_PER_BLOCK;  // 782
    nn_dist_kernel<<<blocks, 256, 0, stream>>>(in_pose, train_poses, best);

    nn_gather_kernel<<<(PREDF + 255) / 256, 256, 0, stream>>>(best, target_vels, out);
}